// GATGNN_18554258718932
// MI455X (gfx1250) — compile-verified
//
#include <hip/hip_runtime.h>
#include <math.h>

typedef float v2f __attribute__((ext_vector_type(2)));
typedef float v8f __attribute__((ext_vector_type(8)));
typedef int   v4i __attribute__((ext_vector_type(4)));

#define NEG_ATT 0.2f
#define NEG_ACT 0.01f

#if defined(__has_builtin)
#if __has_builtin(__builtin_amdgcn_global_load_async_to_lds_b128)
#define HAVE_ASYNC_LDS 1
#endif
#endif
#ifndef HAVE_ASYNC_LDS
#define HAVE_ASYNC_LDS 0
#endif

#if HAVE_ASYNC_LDS
typedef __attribute__((address_space(1))) v4i gas_v4i;   // global int4
typedef __attribute__((address_space(3))) v4i las_v4i;   // LDS int4
#endif

__device__ __forceinline__ float leakyf(float x, float s) { return x > 0.f ? x : s * x; }

// Monotonic float <-> uint mapping so unsigned atomicMax implements float max.
__device__ __forceinline__ unsigned ord_f32(float f) {
  unsigned b = __float_as_uint(f);
  return (b & 0x80000000u) ? ~b : (b | 0x80000000u);
}
__device__ __forceinline__ float deord_f32(unsigned u) {
  return (u & 0x80000000u) ? __uint_as_float(u & 0x7FFFFFFFu) : __uint_as_float(~u);
}

// ---------------------------------------------------------------------------
// Dense GEMM  O[Nrows,Fout] = X[Nrows,Fin] @ W[Fin,Fout]  via V_WMMA_F32_16X16X4_F32.
// Block = 8 waves. blockIdx.y selects a 16-wide column tile of W, staged into LDS
// as Wk[k][c] (natural layout). Staging uses GLOBAL_LOAD_ASYNC_TO_LDS_B128 when
// the tile is full-width (Fout % 16 == 0); the ragged decoder tile (Fout=3) uses
// a guarded synchronous copy with zero padding. Each wave then computes one
// 16-row tile with a branch-free inner loop: global float2 (A) + ds b32 pair (B)
// + wmma. Fragment layouts per CDNA5 ISA 7.12.2.
// Requires Nrows % 16 == 0 (N = 50000 = 3125*16) and Fin % 4 == 0.
// ---------------------------------------------------------------------------
#define MAX_FIN 128

__global__ __launch_bounds__(256)
void gemm16x16x4(const float* __restrict__ X, const float* __restrict__ W,
                 float* __restrict__ O, int Nrows, int Fin, int Fout) {
  __shared__ float Wk[MAX_FIN * 16];   // [k][col_local], 8 KB max
  const int col0 = blockIdx.y << 4;

#if HAVE_ASYNC_LDS
  if ((Fout & 15) == 0) {
    // Async DMA: each selected lane copies one 16B chunk global -> LDS.
    // Tile row k is 16 consecutive floats at W + k*Fout + col0 (16B aligned).
    for (int i = threadIdx.x; i < (Fin << 2); i += 256) {
      const int k  = i >> 2;          // W row
      const int c4 = (i & 3) << 2;    // float offset of this 16B chunk
      const float* g = W + (size_t)k * Fout + col0 + c4;
      __builtin_amdgcn_global_load_async_to_lds_b128(
          (gas_v4i*)(void*)const_cast<float*>(g),
          (las_v4i*)(void*)&Wk[(k << 4) + c4], 0, 0);
    }
#if __has_builtin(__builtin_amdgcn_s_wait_asynccnt)
    __builtin_amdgcn_s_wait_asynccnt(0);
#else
    asm volatile("s_wait_asynccnt 0x0" ::: "memory");
#endif
  } else
#endif
  {
    for (int i = threadIdx.x; i < (Fin << 4); i += 256) {
      const int k = i >> 4;           // W row
      const int c = i & 15;           // local col
      const int col = col0 + c;
      Wk[(k << 4) + c] = (col < Fout) ? W[(size_t)k * Fout + col] : 0.f;
    }
  }
  __syncthreads();

  const int wave = threadIdx.x >> 5;
  const int lane = threadIdx.x & 31;
  const int row0 = (blockIdx.x * 8 + wave) << 4;
  if (row0 >= Nrows) return;

  const int half = lane >> 4;      // 0 or 1
  const int l16  = lane & 15;
  const float* __restrict__ xrow = X + (size_t)(row0 + l16) * Fin + (half << 1);
  const float* wcol = Wk + (half << 5) + l16;   // (k + 2*half)*16 + l16

  v8f acc = {};
#pragma unroll 4
  for (int k = 0; k < Fin; k += 4) {
    v2f a = *(const v2f*)(xrow + k);            // global_load_b64
    v2f b;
    b.x = wcol[(k << 4)];                       // ds loads, 64B apart
    b.y = wcol[(k << 4) + 16];
    // (neg_a, A, neg_b, B, c_mod, C, reuse_a, reuse_b)
    acc = __builtin_amdgcn_wmma_f32_16x16x4_f32(false, a, false, b, (short)0, acc,
                                                false, false);
  }

  const int col = col0 + l16;
  if (col < Fout) {
#pragma unroll
    for (int r = 0; r < 8; ++r) {
      const int m = row0 + r + (half << 3);
      O[(size_t)m * Fout + col] = acc[r];
    }
  }
}

// alpha_s[n,h] = sum_c h[n,h,c]*a_src[h,c] ; same for alpha_d
__global__ __launch_bounds__(256)
void alpha_kernel(const float* __restrict__ hfeat, const float* __restrict__ a_s,
                  const float* __restrict__ a_d, float* __restrict__ als,
                  float* __restrict__ ald, int Nn, int H, int C) {
  const int idx = blockIdx.x * blockDim.x + threadIdx.x;
  if (idx >= Nn * H) return;
  const int n = idx / H, h = idx % H;
  const float* hp = hfeat + (size_t)n * H * C + h * C;
  float s = 0.f, d = 0.f;
  for (int c = 0; c < C; ++c) {
    const float v = hp[c];
    s = fmaf(v, a_s[h * C + c], s);
    d = fmaf(v, a_d[h * C + c], d);
  }
  als[idx] = s;
  ald[idx] = d;
}

__device__ __forceinline__ void edge_ends(const int* __restrict__ ei, int e, int E,
                                          int& s, int& d) {
  if (e < E) { s = ei[e]; d = ei[E + e]; }   // row0 = src, row1 = dst
  else       { s = d = e - E; }              // appended self loops
}

// segment max of leaky-relu'd logits over dst
__global__ __launch_bounds__(256)
void att_max_kernel(const int* __restrict__ ei, const float* __restrict__ als,
                    const float* __restrict__ ald, unsigned* __restrict__ mbits,
                    int E, int Nn, int H) {
  const int idx = blockIdx.x * blockDim.x + threadIdx.x;
  if (idx >= (E + Nn) * H) return;
  const int e = idx / H, h = idx % H;
  int s, d; edge_ends(ei, e, E, s, d);
  const float lg = leakyf(als[s * H + h] + ald[d * H + h], NEG_ATT);
  atomicMax(&mbits[d * H + h], ord_f32(lg));
}

// segment sum of exp(logit - max) over dst
__global__ __launch_bounds__(256)
void exp_sum_kernel(const int* __restrict__ ei, const float* __restrict__ als,
                    const float* __restrict__ ald, const unsigned* __restrict__ mbits,
                    float* __restrict__ denom, int E, int Nn, int H) {
  const int idx = blockIdx.x * blockDim.x + threadIdx.x;
  if (idx >= (E + Nn) * H) return;
  const int e = idx / H, h = idx % H;
  int s, d; edge_ends(ei, e, E, s, d);
  const float lg = leakyf(als[s * H + h] + ald[d * H + h], NEG_ATT);
  const float m  = deord_f32(mbits[d * H + h]);
  atomicAdd(&denom[d * H + h], __expf(lg - m));
}

// out[dst,h,:] += att[e,h] * h[src,h,:]   (one wave per edge)
__global__ __launch_bounds__(256)
void aggregate_kernel(const int* __restrict__ ei, const float* __restrict__ als,
                      const float* __restrict__ ald, const unsigned* __restrict__ mbits,
                      const float* __restrict__ denom, const float* __restrict__ hfeat,
                      float* __restrict__ out, int E, int Nn, int H, int C) {
  const int Fout = H * C;
  const int wave = blockIdx.x * (blockDim.x >> 5) + (threadIdx.x >> 5);
  const int lane = threadIdx.x & 31;
  if (wave >= E + Nn) return;
  int s, d; edge_ends(ei, wave, E, s, d);

  if (Fout == 128 && C == 16 && H == 8) {
    // lanes 0..7 each compute one head's attention coefficient, then broadcast
    float att_mine = 0.f;
    if (lane < 8) {
      const float lg = leakyf(als[s * 8 + lane] + ald[d * 8 + lane], NEG_ATT);
      att_mine = __expf(lg - deord_f32(mbits[d * 8 + lane])) / denom[d * 8 + lane];
    }
    const int   f0  = lane << 2;                 // 4 contiguous features per lane
    const float att = __shfl(att_mine, f0 >> 4); // head = f0 / 16
    const float4 hv = *(const float4*)(hfeat + (size_t)s * 128 + f0);
    float* op = out + (size_t)d * 128 + f0;
    atomicAdd(op + 0, att * hv.x);
    atomicAdd(op + 1, att * hv.y);
    atomicAdd(op + 2, att * hv.z);
    atomicAdd(op + 3, att * hv.w);
  } else {
    for (int f = lane; f < Fout; f += 32) {
      const int h = f / C;
      const float lg  = leakyf(als[s * H + h] + ald[d * H + h], NEG_ATT);
      const float att = __expf(lg - deord_f32(mbits[d * H + h])) / denom[d * H + h];
      atomicAdd(&out[(size_t)d * Fout + f], att * hfeat[(size_t)s * Fout + f]);
    }
  }
}

// out[n,f] = b[f]  (bias pre-load; aggregation accumulates on top)
__global__ __launch_bounds__(256)
void init_bias_kernel(float* __restrict__ out, const float* __restrict__ b,
                      int total, int Fout) {
  const int idx = blockIdx.x * blockDim.x + threadIdx.x;
  if (idx >= total) return;
  out[idx] = b[idx % Fout];
}

// encoder-only post-activation
__global__ __launch_bounds__(256)
void act_kernel(float* __restrict__ out, int total) {
  const int idx = blockIdx.x * blockDim.x + threadIdx.x;
  if (idx >= total) return;
  out[idx] = leakyf(out[idx], NEG_ACT);
}

// ---------------------------------------------------------------------------
extern "C" void kernel_launch(void* const* d_in, const int* in_sizes, int n_in,
                              void* d_out, int out_size, void* d_ws, size_t ws_size,
                              hipStream_t stream) {
  const float* x      = (const float*)d_in[0];
  const int*   ei     = (const int*)  d_in[1];
  const float* enc_W  = (const float*)d_in[2];
  const float* enc_as = (const float*)d_in[3];
  const float* enc_ad = (const float*)d_in[4];
  const float* enc_b  = (const float*)d_in[5];
  const float* hid_W  = (const float*)d_in[6];
  const float* hid_as = (const float*)d_in[7];
  const float* hid_ad = (const float*)d_in[8];
  const float* hid_b  = (const float*)d_in[9];
  const float* dec_W  = (const float*)d_in[10];
  const float* dec_as = (const float*)d_in[11];
  const float* dec_ad = (const float*)d_in[12];
  const float* dec_b  = (const float*)d_in[13];

  const int N  = in_sizes[0] / 16;     // 50000 (multiple of 16)
  const int E  = in_sizes[1] / 2;      // 800000
  const int HC = 128;

  // workspace carve-up
  char* ws = (char*)d_ws;
  float*    bufA  = (float*)ws;    ws += (size_t)N * HC * sizeof(float);  // gemm output h
  float*    bufB  = (float*)ws;    ws += (size_t)N * HC * sizeof(float);  // layer output
  float*    als   = (float*)ws;    ws += (size_t)N * 8 * sizeof(float);
  float*    ald   = (float*)ws;    ws += (size_t)N * 8 * sizeof(float);
  unsigned* mbits = (unsigned*)ws; ws += (size_t)N * 8 * sizeof(unsigned);
  float*    denom = (float*)ws;    ws += (size_t)N * 8 * sizeof(float);

  auto layer = [&](const float* X, int Fin, const float* W, const float* a_s,
                   const float* a_d, const float* b, float* hbuf, float* out,
                   int H, int C, int doAct) {
    const int Fo = H * C;
    dim3 gg((N + 127) / 128, (Fo + 15) / 16);
    gemm16x16x4<<<gg, 256, 0, stream>>>(X, W, hbuf, N, Fin, Fo);
    const int nh = N * H;
    alpha_kernel<<<(nh + 255) / 256, 256, 0, stream>>>(hbuf, a_s, a_d, als, ald, N, H, C);
    (void)hipMemsetAsync(mbits, 0, (size_t)nh * sizeof(unsigned), stream);
    (void)hipMemsetAsync(denom, 0, (size_t)nh * sizeof(float), stream);
    const int nf = N * Fo;
    init_bias_kernel<<<(nf + 255) / 256, 256, 0, stream>>>(out, b, nf, Fo);
    const int tot = (E + N) * H;
    att_max_kernel<<<(tot + 255) / 256, 256, 0, stream>>>(ei, als, ald, mbits, E, N, H);
    exp_sum_kernel<<<(tot + 255) / 256, 256, 0, stream>>>(ei, als, ald, mbits, denom, E, N, H);
    const int waves = E + N;                       // one wave per edge, 8 waves/block
    aggregate_kernel<<<(waves + 7) / 8, 256, 0, stream>>>(ei, als, ald, mbits, denom,
                                                          hbuf, out, E, N, H, C);
    if (doAct)
      act_kernel<<<(nf + 255) / 256, 256, 0, stream>>>(out, nf);
  };

  // encoder: 16 -> 8x16, then leaky_relu(0.01)
  layer(x, 16, enc_W, enc_as, enc_ad, enc_b, bufA, bufB, 8, 16, 1);
  // hidden layers: 128 -> 8x16 (X=bufB is consumed by the GEMM before bufB is rewritten)
  for (int i = 0; i < 3; ++i) {
    layer(bufB, HC, hid_W + (size_t)i * HC * HC, hid_as + (size_t)i * HC,
          hid_ad + (size_t)i * HC, hid_b + (size_t)i * HC, bufA, bufB, 8, 16, 0);
  }
  // decoder: 128 -> 1x3, no activation, straight into d_out
  layer(bufB, HC, dec_W, dec_as, dec_ad, dec_b, bufA, (float*)d_out, 1, 3, 0);
}